// AnalyticalMinkowskiLoss_67053029425874
// MI455X (gfx1250) — compile-verified
//
#include <hip/hip_runtime.h>
#include <math.h>

#define HH 512
#define WW 512
#define BB 4
#define QQ 16
#define NPIX (HH*WW)
#define NTILES (NPIX/256)   /* 1024 tiles of 256 pixels per image */

typedef __attribute__((ext_vector_type(2))) float v2f;
typedef __attribute__((ext_vector_type(8))) float v8f;

__device__ __forceinline__ float sigf(float x) {
    return 1.0f / (1.0f + __expf(-x));
}

// ---------------------------------------------------------------------------
// Separable clamped min/max pass. Clamped (replicate) sampling is equivalent
// to the reference's -inf-padded max-pool (pad value never wins; duplicates
// are harmless for min/max). 3x3 pool = hpass(R=1) + vpass(R=1); 15x15 = R=7.
// ---------------------------------------------------------------------------
template<bool ISMAX, int R, bool HORIZ>
__global__ void morph_pass(const float* __restrict__ in, float* __restrict__ out) {
    int idx = blockIdx.x * 256 + threadIdx.x;          // over B*H*W
    int b   = idx / NPIX;
    int rem = idx - b * NPIX;
    int i   = rem >> 9;
    int j   = rem & (WW - 1);
    const float* p = in + b * NPIX;
    float v = ISMAX ? -3.402823466e38f : 3.402823466e38f;
    #pragma unroll
    for (int d = -R; d <= R; ++d) {
        int ii = i, jj = j;
        if (HORIZ) { jj = j + d; jj = jj < 0 ? 0 : (jj > WW-1 ? WW-1 : jj); }
        else       { ii = i + d; ii = ii < 0 ? 0 : (ii > HH-1 ? HH-1 : ii); }
        float x = p[ii * WW + jj];
        v = ISMAX ? fmaxf(v, x) : fminf(v, x);
    }
    out[idx] = v;
}

// wave32 shuffle reduce + cross-wave LDS; result valid in thread 0.
__device__ __forceinline__ float blockSum256(float v, float* lds) {
    int tid = threadIdx.x;
    #pragma unroll
    for (int off = 16; off > 0; off >>= 1) v += __shfl_down(v, off, 32);
    __syncthreads();                       // protect lds from previous call
    if ((tid & 31) == 0) lds[tid >> 5] = v;
    __syncthreads();
    float r = 0.f;
    if (tid == 0) {
        r = lds[0];
        #pragma unroll
        for (int k = 1; k < 8; ++k) r += lds[k];
    }
    return r;
}

// ---------------------------------------------------------------------------
// Fused per-pixel metrics: for each of 16 thresholds compute soft occupancy
// (area), central-difference perimeter (replicate pad == clamped neighbor),
// and the 4 Euler-characteristic terms on the topologically cleaned field.
// Emits per-tile partial sums: part[b][metric][tile][q].
// ---------------------------------------------------------------------------
__global__ void fused_metrics(const float* __restrict__ pred,
                              const float* __restrict__ fieldT,
                              const float* __restrict__ locmax,
                              float* __restrict__ part) {
    __shared__ float lds[8];
    int tid  = threadIdx.x;
    int tile = blockIdx.x;                 // 0..1023
    int b    = blockIdx.y;                 // 0..3
    int p    = tile * 256 + tid;
    int i    = p >> 9;
    int j    = p & (WW - 1);

    const float* P  = pred   + b * NPIX;
    const float* FT = fieldT + b * NPIX;
    const float* LM = locmax + b * NPIX;

    int jm = j > 0 ? j - 1 : 0;
    int jp = j < WW-1 ? j + 1 : WW-1;
    int ip = i < HH-1 ? i + 1 : HH-1;
    int im = i > 0 ? i - 1 : 0;

    float pc = P[i*WW + j],  pl = P[i*WW + jm], pr = P[i*WW + jp];
    float pu = P[im*WW + j], pd = P[ip*WW + j];
    float f00 = FT[i*WW + j],  f01 = FT[i*WW + jp];
    float f10 = FT[ip*WW + j], f11 = FT[ip*WW + jp];
    float l00 = LM[i*WW + j],  l01 = LM[i*WW + jp];
    float l10 = LM[ip*WW + j], l11 = LM[ip*WW + jp];
    bool hasR = (j < WW - 1), hasD = (i < HH - 1);

    for (int q = 0; q < QQ; ++q) {
        float th   = (float)q + 0.5f;
        float invT = 10.0f / th;                       // temps = 0.1*th (>= 0.001 always)
        float thp  = th + 1.8699999839067458f;         // persistence threshold

        // area + perimeter from p_raw(pred)
        float s_c = sigf((pc - th) * invT);
        float s_l = sigf((pl - th) * invT);
        float s_r = sigf((pr - th) * invT);
        float s_u = sigf((pu - th) * invT);
        float s_d = sigf((pd - th) * invT);
        float dx = 0.5f * (s_r - s_l);
        float dy = 0.5f * (s_d - s_u);
        float per = sqrtf(dx*dx + dy*dy + 1e-8f);

        // p_topo = min(sigmoid(field_topo), sigmoid(local_max persistence))
        float pt00 = fminf(sigf((f00 - th) * invT), sigf((l00 - thp) * invT));
        float pt01 = fminf(sigf((f01 - th) * invT), sigf((l01 - thp) * invT));
        float pt10 = fminf(sigf((f10 - th) * invT), sigf((l10 - thp) * invT));
        float pt11 = fminf(sigf((f11 - th) * invT), sigf((l11 - thp) * invT));

        float ex = hasR ? fminf(pt00, pt01) : 0.f;
        float ey = hasD ? fminf(pt00, pt10) : 0.f;
        float ff = (hasR && hasD)
                 ? fminf(fminf(pt00, pt01), fminf(pt10, pt11)) : 0.f;

        float vals[6] = { s_c, per, pt00, ex, ey, ff };
        #pragma unroll
        for (int m = 0; m < 6; ++m) {
            float s = blockSum256(vals[m], lds);
            if (tid == 0)
                part[(((size_t)(b*6 + m) * NTILES + tile) * QQ) + q] = s;
        }
    }
}

// ---------------------------------------------------------------------------
// Finalize (1 wave, EXEC all ones): WMMA column-sum reduction.
// C += ones(16x4) x B(4x16) with V_WMMA_F32_16X16X4_F32; columns are the 16
// thresholds, k-rows are 4 consecutive tile partials per step. Any bijective
// k-slot mapping of the 4 loaded rows still sums the column correctly.
// Lane n (n<16) of C VGPR0 then holds the total for q=n.
// ---------------------------------------------------------------------------
__global__ void finalize_kernel(const float* __restrict__ part,
                                const float* __restrict__ tgt,
                                float* __restrict__ out) {
    int lane = threadIdx.x;                // 0..31, full wave
    int qi   = lane & 15;
    int r0   = (lane < 16) ? 0 : 2;
    v2f ones = { 1.0f, 1.0f };
    float loss = 0.f;

    for (int b = 0; b < BB; ++b) {
        float met[6];
        #pragma unroll
        for (int m = 0; m < 6; ++m) {
            v8f c = {};
            const float* base = part + (size_t)(b*6 + m) * NTILES * QQ;
            for (int t = 0; t < NTILES; t += 4) {
                v2f bm;
                bm[0] = base[(t + r0    ) * QQ + qi];
                bm[1] = base[(t + r0 + 1) * QQ + qi];
                c = __builtin_amdgcn_wmma_f32_16x16x4_f32(
                        false, ones, false, bm, (short)0, c, false, false);
            }
            met[m] = c[0];                 // col-sum for q = lane (lanes 0..15)
        }

        float area = met[0] * 4.0f;                    // PIXEL_AREA
        float perim = met[1] * 2.0f;                   // PIXEL_SIZE_KM
        float euler = met[2] - met[3] - met[4] + met[5];
        float pg[3] = { area, perim, euler };

        float tr[4];
        #pragma unroll
        for (int c4 = 0; c4 < 4; ++c4) {
            float t = tgt[(b*4 + c4) * QQ + qi];
            tr[c4] = copysignf(expm1f(fabsf(t)), t);
        }
        float tp[3] = { tr[0], tr[1], tr[2] - tr[3] };

        #pragma unroll
        for (int cdim = 0; cdim < 3; ++cdim) {
            float plog = copysignf(log1pf(fabsf(pg[cdim])), pg[cdim]);
            float tlog = copysignf(log1pf(fabsf(tp[cdim])), tp[cdim]);
            float ad   = fabsf(plog - tlog);
            float adm  = (lane < 16) ? ad : 0.f;
            float tot  = adm;
            #pragma unroll
            for (int off = 16; off > 0; off >>= 1) tot += __shfl_down(tot, off, 32);
            float ad0  = __shfl(ad, 0, 32);
            float ad15 = __shfl(ad, 15, 32);
            // trapezoid with dth == 1: sum - 0.5*(first+last)
            if (lane == 0) loss += tot - 0.5f * (ad0 + ad15);
        }
    }
    if (lane == 0) out[0] = loss * 0.25f;              // mean over batch
}

extern "C" void kernel_launch(void* const* d_in, const int* in_sizes, int n_in,
                              void* d_out, int out_size, void* d_ws, size_t ws_size,
                              hipStream_t stream) {
    (void)in_sizes; (void)n_in; (void)out_size; (void)ws_size;
    const float* pred = (const float*)d_in[0];   // [4,1,512,512]
    const float* tgt  = (const float*)d_in[1];   // [4,4,16]
    float* out = (float*)d_out;
    float* w   = (float*)d_ws;

    size_t N = (size_t)BB * NPIX;
    float* A    = w;                 // scratch ping
    float* Bb   = w + N;             // scratch pong
    float* C    = w + 2*N;           // field_topo
    float* D    = w + 3*N;           // local_max
    float* part = w + 4*N;           // [4][6][1024][16] tile partials

    dim3 blk(256);
    dim3 g1((unsigned)((BB * NPIX) / 256));

    // dilate3 -> erode3 -> erode3 -> dilate3 (each separable h+v)
    morph_pass<true, 1, true ><<<g1, blk, 0, stream>>>(pred, A);
    morph_pass<true, 1, false><<<g1, blk, 0, stream>>>(A, Bb);   // dilated
    morph_pass<false,1, true ><<<g1, blk, 0, stream>>>(Bb, A);
    morph_pass<false,1, false><<<g1, blk, 0, stream>>>(A, Bb);   // closed
    morph_pass<false,1, true ><<<g1, blk, 0, stream>>>(Bb, A);
    morph_pass<false,1, false><<<g1, blk, 0, stream>>>(A, Bb);   // eroded
    morph_pass<true, 1, true ><<<g1, blk, 0, stream>>>(Bb, A);
    morph_pass<true, 1, false><<<g1, blk, 0, stream>>>(A, C);    // field_topo
    // 15x15 max on field_topo
    morph_pass<true, 7, true ><<<g1, blk, 0, stream>>>(C, A);
    morph_pass<true, 7, false><<<g1, blk, 0, stream>>>(A, D);    // local_max

    fused_metrics<<<dim3(NTILES, BB), blk, 0, stream>>>(pred, C, D, part);
    finalize_kernel<<<1, 32, 0, stream>>>(part, tgt, out);
}